// SimplifiedContinuousMambaBlock_2851858285244
// MI455X (gfx1250) — compile-verified
//
#include <hip/hip_runtime.h>
#include <hip/hip_bf16.h>

// Problem constants (match reference)
#define Bb      4
#define LL      2048
#define DIN     256
#define HH      512
#define DINNER  1024
#define DSTATE  16
#define DCONV   4
#define DTRANK  32
#define TDIM    256
#define NROWS   (Bb * LL)        // 8192

#define KC      32               // K-chunk staged in LDS per stage
#define SW      36               // padded LDS row stride (floats): conflict-free + 16B aligned

typedef __attribute__((ext_vector_type(2))) float v2f;
typedef __attribute__((ext_vector_type(8))) float v8f;

#if __has_builtin(__builtin_amdgcn_global_load_async_to_lds_b128) && \
    __has_builtin(__builtin_amdgcn_s_wait_asynccnt)
#define ASYNC_LDS 1
typedef int v4i_ __attribute__((vector_size(16)));
typedef __attribute__((address_space(1))) v4i_ gv4i;   // global (AS1) int32x4
typedef __attribute__((address_space(3))) v4i_ lv4i;   // LDS    (AS3) int32x4
#else
#define ASYNC_LDS 0
#endif

// ---------------------------------------------------------------------------
// fp32 WMMA GEMM:  Cout[m][n] = op( sum_k A[m][k]*W[n][k] + bias[n] + Cin[m][n] )
// A: (M x K) row-major, lda. W: (N x K) row-major (weights as stored by reference).
// Block tile: 128(M) x 64(N); 8 waves share one N-group, W k-slices double-
// buffered through LDS (async-to-LDS when available). Wave tile: 16x64, K step 4
// via V_WMMA_F32_16X16X4_F32. op: 0 = identity, 1 = softplus.
// ---------------------------------------------------------------------------
__global__ __launch_bounds__(256) void wmma_gemm_f32(
    const float* __restrict__ A, int lda,
    const float* __restrict__ W,
    const float* __restrict__ bias,
    const float* __restrict__ Cin,
    float* __restrict__ Cout, int ldc,
    int M, int N, int K, int op)
{
    __shared__ float ldsW[2][64 * SW];      // 2 x 9216 B

    const int tid  = threadIdx.x;
    const int ln   = tid & 31;
    const int wave = tid >> 5;

    const int nGroups = N >> 6;             // N / 64
    const int bm = blockIdx.x / nGroups;    // 128-row supertile
    const int ng = blockIdx.x % nGroups;
    const int m0 = bm * 128 + wave * 16;
    const int n0 = ng << 6;

    const int mlane = ln & 15;              // M index (A frag), N index (B/C frags)
    const int khalf = ln >> 4;              // K pair select {0,1} vs {2,3}

    // Accumulator init (residual / accumulate path); uniform branch keeps EXEC full
    v8f acc[4];
    if (Cin) {
#pragma unroll
        for (int t = 0; t < 4; ++t) {
            const int n = n0 + t * 16 + mlane;
#pragma unroll
            for (int r = 0; r < 8; ++r)
                acc[t][r] = Cin[(size_t)(m0 + r + 8 * khalf) * ldc + n];
        }
    } else {
#pragma unroll
        for (int t = 0; t < 4; ++t)
#pragma unroll
            for (int r = 0; r < 8; ++r) acc[t][r] = 0.f;
    }

    // A fragment pointer: a[j] = A[m0+mlane][k + 2*khalf + j]  (ISA 16x4 f32 layout)
    const float* Ap = A + (size_t)(m0 + mlane) * lda + 2 * khalf;

    // Cooperative W fill: thread -> (row fn of 64, 8 floats at fk)
    const int fn = tid >> 2;                // 0..63
    const int fk = (tid & 3) * 8;           // 0,8,16,24
    const float* Wfill = W + (size_t)(n0 + fn) * K + fk;

    auto fill_stage = [&](int buf, int kbase) {
        const float* src = Wfill + kbase;
        float* dst = &ldsW[buf][fn * SW + fk];
#if ASYNC_LDS
        __builtin_amdgcn_global_load_async_to_lds_b128(
            (gv4i*)(void*)src,       (lv4i*)(void*)dst,       0, 0);
        __builtin_amdgcn_global_load_async_to_lds_b128(
            (gv4i*)(void*)(src + 4), (lv4i*)(void*)(dst + 4), 0, 0);
#else
        float4 f0 = *(const float4*)(src);
        float4 f1 = *(const float4*)(src + 4);
        *(float4*)dst = f0;
        *(float4*)(dst + 4) = f1;
#endif
    };

    const int nChunks = K / KC;

    fill_stage(0, 0);
#if ASYNC_LDS
    __builtin_amdgcn_s_wait_asynccnt(0);
#endif
    __syncthreads();

    for (int c = 0; c < nChunks; ++c) {
        const int kbase = c * KC;
        if (c + 1 < nChunks) {
            fill_stage((c + 1) & 1, kbase + KC);            // overlap next W slice
            __builtin_prefetch(Ap + kbase + KC, 0, 3);      // keep A stream warm in WGP$
        }
        const float* lw = ldsW[c & 1];

        // Batch all A fragments for this chunk: one load-wait amortized over 32 WMMAs
        v2f areg[8];
#pragma unroll
        for (int s = 0; s < 8; ++s)
            areg[s] = *(const v2f*)(Ap + kbase + 4 * s);

#pragma unroll
        for (int s = 0; s < 8; ++s) {
            const int kk = 4 * s + 2 * khalf;
            v2f b0 = *(const v2f*)(lw + (0 * 16 + mlane) * SW + kk);
            v2f b1 = *(const v2f*)(lw + (1 * 16 + mlane) * SW + kk);
            v2f b2 = *(const v2f*)(lw + (2 * 16 + mlane) * SW + kk);
            v2f b3 = *(const v2f*)(lw + (3 * 16 + mlane) * SW + kk);
            acc[0] = __builtin_amdgcn_wmma_f32_16x16x4_f32(false, areg[s], false, b0, (short)0, acc[0], false, false);
            acc[1] = __builtin_amdgcn_wmma_f32_16x16x4_f32(false, areg[s], false, b1, (short)0, acc[1], false, false);
            acc[2] = __builtin_amdgcn_wmma_f32_16x16x4_f32(false, areg[s], false, b2, (short)0, acc[2], false, false);
            acc[3] = __builtin_amdgcn_wmma_f32_16x16x4_f32(false, areg[s], false, b3, (short)0, acc[3], false, false);
        }
#if ASYNC_LDS
        __builtin_amdgcn_s_wait_asynccnt(0);
#endif
        __syncthreads();
    }

    // Epilogue: bias + optional softplus (C/D layout: n = ln&15, m = r + 8*(ln>>4))
#pragma unroll
    for (int t = 0; t < 4; ++t) {
        const int n = n0 + t * 16 + mlane;
        const float bv = bias ? bias[n] : 0.f;
#pragma unroll
        for (int r = 0; r < 8; ++r) {
            float v = acc[t][r] + bv;
            if (op == 1) v = (v > 20.f) ? v : log1pf(__expf(v));
            Cout[(size_t)(m0 + r + 8 * khalf) * ldc + n] = v;
        }
    }
}

// ---------------------------------------------------------------------------
// Time encoder features: periodic[b,l, 0:128] = sin(dt*freq+phase), [128:256] = cos
// ---------------------------------------------------------------------------
__global__ __launch_bounds__(256) void time_enc_kernel(
    const float* __restrict__ ts, const float* __restrict__ freq,
    const float* __restrict__ phase, float* __restrict__ P)
{
    const int idx = blockIdx.x * 256 + threadIdx.x;   // NROWS*128 threads
    const int j  = idx & 127;
    const int bl = idx >> 7;
    const int l  = bl & (LL - 1);
    const float t  = ts[bl];
    const float td = (l == 0) ? 0.f : (t - ts[bl - 1]);
    const float ang = td * freq[j] + phase[j];
    float s, c;
    __sincosf(ang, &s, &c);
    P[(size_t)bl * TDIM + j]       = s;
    P[(size_t)bl * TDIM + 128 + j] = c;
}

// ---------------------------------------------------------------------------
// Depthwise causal conv1d (k=4) + SiLU.  x lives in xz[..., 0:DINNER].
// ---------------------------------------------------------------------------
__global__ __launch_bounds__(256) void conv_silu_kernel(
    const float* __restrict__ xz, const float* __restrict__ cw,
    const float* __restrict__ cb, float* __restrict__ xc)
{
    const int idx = blockIdx.x * 256 + threadIdx.x;   // NROWS*DINNER threads
    const int d  = idx & (DINNER - 1);
    const int bl = idx >> 10;
    const int l  = bl & (LL - 1);
    float acc = cb[d];
#pragma unroll
    for (int k = 0; k < DCONV; ++k) {
        const int ls = l - (DCONV - 1) + k;
        if (ls >= 0)
            acc = fmaf(cw[d * DCONV + k], xz[(size_t)(bl - (DCONV - 1) + k) * (2 * DINNER) + d], acc);
    }
    xc[(size_t)bl * DINNER + d] = acc / (1.f + __expf(-acc));   // SiLU
}

// ---------------------------------------------------------------------------
// Selective scan. One lane per (b,d): 16 states in VGPRs; B/C staged in LDS
// 64 timesteps at a time. Fuses + x*D and *silu(z); writes y in-place into xc.
// dt was written over the x-half of xz; z is the upper half of xz.
// ---------------------------------------------------------------------------
__global__ __launch_bounds__(256) void scan_kernel(
    const float* __restrict__ xz, float* __restrict__ xc,
    const float* __restrict__ xdbl, const float* __restrict__ A_log,
    const float* __restrict__ Dv)
{
    const int b = blockIdx.x >> 2;
    const int d = ((blockIdx.x & 3) << 8) + threadIdx.x;
    __shared__ float bc[64 * 32];                     // 64 steps x (B[16] | C[16])

    float Ar[DSTATE], h[DSTATE];
#pragma unroll
    for (int n = 0; n < DSTATE; ++n) {
        Ar[n] = -__expf(A_log[(size_t)d * DSTATE + n]);
        h[n]  = 0.f;
    }
    const float Dd = Dv[d];
    const size_t base = (size_t)b * LL;

    for (int l0 = 0; l0 < LL; l0 += 64) {
        __syncthreads();
        for (int i = threadIdx.x; i < 64 * 32; i += 256) {
            const int s = i >> 5, n = i & 31;
            bc[i] = xdbl[(base + l0 + s) * 64 + DTRANK + n];   // cols 32..63 = B|C
        }
        __syncthreads();
        for (int s = 0; s < 64; ++s) {
            const size_t row = base + l0 + s;
            const float dtv = xz[row * (2 * DINNER) + d];
            const float xv  = xc[row * DINNER + d];
            const float zv  = xz[row * (2 * DINNER) + DINNER + d];
            const float dtx = dtv * xv;
            float y = 0.f;
#pragma unroll
            for (int n = 0; n < DSTATE; ++n) {
                const float dA = __expf(dtv * Ar[n]);
                h[n] = fmaf(dA, h[n], dtx * bc[s * 32 + n]);
                y = fmaf(h[n], bc[s * 32 + 16 + n], y);
            }
            y = fmaf(xv, Dd, y);
            xc[row * DINNER + d] = y * (zv / (1.f + __expf(-zv)));   // * silu(z)
        }
    }
}

// ---------------------------------------------------------------------------
// In-place LayerNorm over H=512. One wave32 per row, shfl_xor reductions.
// ---------------------------------------------------------------------------
__global__ __launch_bounds__(256) void ln_kernel(
    float* __restrict__ out, const float* __restrict__ gamma,
    const float* __restrict__ beta)
{
    const int ln   = threadIdx.x & 31;
    const int wave = threadIdx.x >> 5;
    const int row  = blockIdx.x * 8 + wave;
    float* p = out + (size_t)row * HH;

    float v[16];
    float s = 0.f;
#pragma unroll
    for (int i = 0; i < 16; ++i) { v[i] = p[i * 32 + ln]; s += v[i]; }
#pragma unroll
    for (int m = 16; m; m >>= 1) s += __shfl_xor(s, m, 32);
    const float mean = s * (1.f / HH);

    float sq = 0.f;
#pragma unroll
    for (int i = 0; i < 16; ++i) { const float dd = v[i] - mean; sq = fmaf(dd, dd, sq); }
#pragma unroll
    for (int m = 16; m; m >>= 1) sq += __shfl_xor(sq, m, 32);
    const float rstd = rsqrtf(sq * (1.f / HH) + 1e-5f);

#pragma unroll
    for (int i = 0; i < 16; ++i) {
        const int c = i * 32 + ln;
        p[c] = (v[i] - mean) * rstd * gamma[c] + beta[c];
    }
}

// ---------------------------------------------------------------------------
// Host: launch pipeline. Workspace layout (floats):
//   P     [NROWS*TDIM]       periodic features
//   CMB   [NROWS*HH]         combined (u_proj + tf)
//   XZ    [NROWS*2*DINNER]   xz; x-half later overwritten by dt
//   XC    [NROWS*DINNER]     conv+silu x; later overwritten by gated y
//   XDBL  [NROWS*64]         dt_r | B | C
// ---------------------------------------------------------------------------
static inline void launch_gemm(hipStream_t st,
                               const float* A, int lda, const float* W,
                               const float* bias, const float* Cin,
                               float* Cout, int ldc, int M, int N, int K, int op)
{
    const int blocks = (M / 128) * (N / 64);   // all shapes here divide evenly
    wmma_gemm_f32<<<blocks, 256, 0, st>>>(A, lda, W, bias, Cin, Cout, ldc, M, N, K, op);
}

extern "C" void kernel_launch(void* const* d_in, const int* in_sizes, int n_in,
                              void* d_out, int out_size, void* d_ws, size_t ws_size,
                              hipStream_t stream) {
    const float* time_emb  = (const float*)d_in[0];
    const float* timestamps= (const float*)d_in[1];
    const float* W_in      = (const float*)d_in[2];
    const float* b_in      = (const float*)d_in[3];
    const float* freq      = (const float*)d_in[4];
    const float* phase     = (const float*)d_in[5];
    const float* W_te      = (const float*)d_in[6];
    const float* b_te      = (const float*)d_in[7];
    const float* W_inproj  = (const float*)d_in[8];
    const float* conv_w    = (const float*)d_in[9];
    const float* conv_b    = (const float*)d_in[10];
    const float* W_xproj   = (const float*)d_in[11];
    const float* W_dtproj  = (const float*)d_in[12];
    const float* b_dtproj  = (const float*)d_in[13];
    const float* A_log     = (const float*)d_in[14];
    const float* Dvec      = (const float*)d_in[15];
    const float* W_outproj = (const float*)d_in[16];
    const float* gamma     = (const float*)d_in[17];
    const float* beta      = (const float*)d_in[18];
    float* out = (float*)d_out;

    float* ws   = (float*)d_ws;
    float* P    = ws;                                  // 2,097,152
    float* CMB  = P   + (size_t)NROWS * TDIM;          // 4,194,304
    float* XZ   = CMB + (size_t)NROWS * HH;            // 16,777,216
    float* XC   = XZ  + (size_t)NROWS * 2 * DINNER;    // 8,388,608
    float* XDBL = XC  + (size_t)NROWS * DINNER;        //   524,288

    // 1) time-encoder periodic features
    time_enc_kernel<<<(NROWS * 128) / 256, 256, 0, stream>>>(timestamps, freq, phase, P);

    // 2) u_proj = X @ W_in^T + b_in  -> combined
    launch_gemm(stream, time_emb, DIN, W_in, b_in, nullptr, CMB, HH, NROWS, HH, DIN, 0);

    // 3) combined[:, :256] += P @ W_te^T + b_te   (padding: upper 256 cols untouched)
    launch_gemm(stream, P, TDIM, W_te, b_te, CMB, CMB, HH, NROWS, TDIM, TDIM, 0);

    // 4) xz = combined @ W_inproj^T
    launch_gemm(stream, CMB, HH, W_inproj, nullptr, nullptr, XZ, 2 * DINNER, NROWS, 2 * DINNER, HH, 0);

    // 5) depthwise causal conv + SiLU on x-half
    conv_silu_kernel<<<(NROWS * DINNER) / 256, 256, 0, stream>>>(XZ, conv_w, conv_b, XC);

    // 6) x_dbl = x @ W_xproj^T   (dt_r | B | C)
    launch_gemm(stream, XC, DINNER, W_xproj, nullptr, nullptr, XDBL, 64, NROWS, 64, DINNER, 0);

    // 7) dt = softplus(dt_r @ W_dtproj^T + b_dtproj) -> overwrite consumed x-half of XZ
    launch_gemm(stream, XDBL, 64, W_dtproj, b_dtproj, nullptr, XZ, 2 * DINNER, NROWS, DINNER, DTRANK, 1);

    // 8) selective scan, fused + x*D and * silu(z); y written in-place into XC
    scan_kernel<<<Bb * 4, 256, 0, stream>>>(XZ, XC, XDBL, A_log, Dvec);

    // 9) out = y @ W_outproj^T + combined   (residual via Cin)
    launch_gemm(stream, XC, DINNER, W_outproj, nullptr, CMB, out, HH, NROWS, HH, DINNER, 0);

    // 10) LayerNorm in place on d_out
    ln_kernel<<<NROWS / 8, 256, 0, stream>>>(out, gamma, beta);
}